// NTM_46892452938208
// MI455X (gfx1250) — compile-verified
//
#include <hip/hip_runtime.h>

#define NTM_B 2048
#define NTM_T 64
#define NTM_I 9
#define NTM_N 128
#define NTM_V 20
#define NTM_C 100
#define NTM_O 8
#define BB    16
#define EPSF  1e-8f

// padded LDS row strides (even -> keeps v2f 8B alignment; !=pow2 -> fewer bank conflicts)
#define LDXR  34   // xr rows, K=32 used
#define LDH   114  // h rows, K=112 used
#define LDHR  130  // hr rows, K=128 used
#define LDWCT 34   // WcT rows (k-dim 32 used)
#define LDWRT 114  // WrT rows (k-dim 112 used)
#define LDWWT 114  // WwT rows (k-dim 112 used)
#define LDWFT 130  // WfT rows (k-dim 128 used)

typedef float v2f __attribute__((ext_vector_type(2)));
typedef float v8f __attribute__((ext_vector_type(8)));

// ---- fast scalar ops on CDNA5 trans units (guarded so compile can never break) ----
__device__ __forceinline__ float tanh_(float x) {
#if __has_builtin(__builtin_amdgcn_tanhf)
  return __builtin_amdgcn_tanhf(x);   // gfx1250 V_TANH_F32
#else
  return tanhf(x);
#endif
}
__device__ __forceinline__ float rcp_(float x) {
#if __has_builtin(__builtin_amdgcn_rcpf)
  return __builtin_amdgcn_rcpf(x);    // v_rcp_f32
#else
  return 1.f / x;
#endif
}
__device__ __forceinline__ float sqrt_(float x) {
#if __has_builtin(__builtin_amdgcn_sqrtf)
  return __builtin_amdgcn_sqrtf(x);   // v_sqrt_f32
#else
  return sqrtf(x);
#endif
}
__device__ __forceinline__ float sigmoidf_(float x) { return rcp_(1.f + __expf(-x)); }
__device__ __forceinline__ float softplusf_(float x) { return (x > 20.f) ? x : log1pf(__expf(x)); }

// One 16x16 f32 tile: A (row-major [16][ldA], LDS) x B (transposed [N][ldB] "BT[n][k]", LDS).
// V_WMMA_F32_16X16X4_F32 layouts (ISA 7.12.2): per k-step both fragments are two
// K-consecutive floats per lane -> single aligned ds_load_b64 each.
__device__ __forceinline__ v8f gemm_tile_lds(const float* __restrict__ Alds, int ldA,
                                             const float* __restrict__ BTlds, int ldB,
                                             int Kpad, int n0, int lane) {
  const int half = lane >> 4, lr = lane & 15;
  const float* arow = Alds + lr * ldA;
  const float* brow = BTlds + (n0 + lr) * ldB;
  v8f acc = {};
#pragma unroll
  for (int k0 = 0; k0 < Kpad; k0 += 4) {
    const int ka = k0 + 2 * half;
    v2f a = *(const v2f*)(arow + ka);
    v2f b = *(const v2f*)(brow + ka);
    acc = __builtin_amdgcn_wmma_f32_16x16x4_f32(false, a, false, b, (short)0, acc,
                                                false, false);
  }
  return acc;
}

// NTM addressing: content softmax -> interpolate -> circular conv -> sharpen -> renorm.
// 128 threads: thread = (j in 0..15, p in 0..7), each p owns 16 of the 128 memory rows.
__device__ void ntm_address_(const float* __restrict__ kh, const float* __restrict__ beta,
                             const float* __restrict__ g, const float* __restrict__ sv,
                             const float* __restrict__ gamma, const float* __restrict__ normk,
                             const float* __restrict__ wprev, const float* __restrict__ Ml,
                             float* __restrict__ simb, float* __restrict__ red,
                             float* __restrict__ jt, float* __restrict__ wout, int tid) {
  const int j = tid >> 3, p = tid & 7, n0 = p * 16;
  const float nk = normk[j];
  const float bet = beta[j];
  float lmax = -3.0e38f;
  for (int n = n0; n < n0 + 16; ++n) {
    const float* Mr = Ml + ((size_t)j * NTM_N + n) * NTM_V;
    float dot = 0.f, nm = 0.f;
#pragma unroll
    for (int v = 0; v < NTM_V; ++v) {
      float m = Mr[v];
      dot += kh[j * NTM_V + v] * m;
      nm += m * m;
    }
    float sim = bet * dot * rcp_(nk * sqrt_(nm) + EPSF);
    simb[j * NTM_N + n] = sim;
    lmax = fmaxf(lmax, sim);
  }
  red[j * 8 + p] = lmax;
  __syncthreads();
  if (p == 0) {
    float m = red[j * 8];
    for (int q = 1; q < 8; ++q) m = fmaxf(m, red[j * 8 + q]);
    jt[j] = m;
  }
  __syncthreads();
  const float mx = jt[j];
  float lsum = 0.f;
  for (int n = n0; n < n0 + 16; ++n) {
    float e = __expf(simb[j * NTM_N + n] - mx);
    simb[j * NTM_N + n] = e;
    lsum += e;
  }
  red[j * 8 + p] = lsum;
  __syncthreads();
  if (p == 0) {
    float s = 0.f;
    for (int q = 0; q < 8; ++q) s += red[j * 8 + q];
    jt[j] = s;
  }
  __syncthreads();
  const float inv = rcp_(jt[j]);
  const float gg = g[j];
  for (int n = n0; n < n0 + 16; ++n) {
    float wc = simb[j * NTM_N + n] * inv;
    simb[j * NTM_N + n] = gg * wc + (1.f - gg) * wprev[j * NTM_N + n];  // w_g
  }
  __syncthreads();
  const float s0 = sv[j * 3 + 0], s1 = sv[j * 3 + 1], s2 = sv[j * 3 + 2], gm = gamma[j];
  lsum = 0.f;
  for (int n = n0; n < n0 + 16; ++n) {
    float ws = s0 * simb[j * NTM_N + ((n + 1) & (NTM_N - 1))]
             + s1 * simb[j * NTM_N + n]
             + s2 * simb[j * NTM_N + ((n + NTM_N - 1) & (NTM_N - 1))];
    float wp = __powf(fmaxf(ws, 0.f), gm);
    wout[j * NTM_N + n] = wp;
    lsum += wp;
  }
  red[j * 8 + p] = lsum;
  __syncthreads();
  if (p == 0) {
    float s = 0.f;
    for (int q = 0; q < 8; ++q) s += red[j * 8 + q];
    jt[j] = s;
  }
  __syncthreads();
  const float invs = rcp_(jt[j] + EPSF);
  for (int n = n0; n < n0 + 16; ++n) wout[j * NTM_N + n] *= invs;
  __syncthreads();
}

__global__ __launch_bounds__(128) void ntm_fused_kernel(
    const float* __restrict__ x, const float* __restrict__ Wc, const float* __restrict__ bc,
    const float* __restrict__ Wr, const float* __restrict__ br,
    const float* __restrict__ Ww, const float* __restrict__ bw,
    const float* __restrict__ Wf, const float* __restrict__ bf,
    const float* __restrict__ rb, const float* __restrict__ Mb,
    float* __restrict__ out) {
  // ~290 KB of LDS: all recurrent state + transposed zero-padded weights, resident for
  // the entire T=64 scan (only CDNA5's 320KB WGP LDS makes this single-kernel plan work).
  __shared__ float M_l[BB * NTM_N * NTM_V];                       // 160 KB NTM memory
  __shared__ __attribute__((aligned(16))) float xr_l[BB * LDXR];  // [x_t, r_prev]
  __shared__ __attribute__((aligned(16))) float h_l[BB * LDH];    // controller
  __shared__ __attribute__((aligned(16))) float hr_l[BB * LDHR];  // [h, r]
  __shared__ __attribute__((aligned(16))) float WcT_l[LDWCT * 112];  // BT[n][k] weights
  __shared__ __attribute__((aligned(16))) float WrT_l[LDWRT * 32];
  __shared__ __attribute__((aligned(16))) float WwT_l[LDWWT * 80];
  __shared__ __attribute__((aligned(16))) float WfT_l[LDWFT * 16];
  __shared__ float pr_l[BB * 32];
  __shared__ float pw_l[BB * 80];
  __shared__ float r_l[BB * NTM_V];
  __shared__ float wprev_l[BB * NTM_N];  // carries w_w across timesteps
  __shared__ float wr_l[BB * NTM_N];
  __shared__ float sim_l[BB * NTM_N];
  __shared__ float red_l[BB * 8];
  __shared__ float jt_l[BB];
  __shared__ float krh[BB * NTM_V], kwh[BB * NTM_V], eh[BB * NTM_V], ah[BB * NTM_V];
  __shared__ float normkr[BB], normkw[BB];
  __shared__ float betar[BB], gr[BB], gammar[BB], srh[BB * 3];
  __shared__ float betaw[BB], gw[BB], gammaw[BB], srw[BB * 3];
  __shared__ float bc_l[112], br_l[32], bw_l[80], bf_l[16];

  const int tid = threadIdx.x;
  const int lane = tid & 31;
  const int wave = tid >> 5;
  const int bb0 = blockIdx.x * BB;

  // ---- stage weights into LDS: transposed BT[n][k], zero-padded (one time) ----
  for (int idx = tid; idx < 112 * LDWCT; idx += 128) {
    int n = idx / LDWCT, k = idx % LDWCT;
    WcT_l[idx] = (n < NTM_C && k < NTM_I + NTM_V) ? Wc[k * NTM_C + n] : 0.f;
  }
  for (int idx = tid; idx < 32 * LDWRT; idx += 128) {
    int n = idx / LDWRT, k = idx % LDWRT;
    WrT_l[idx] = (n < 26 && k < NTM_C) ? Wr[k * 26 + n] : 0.f;
  }
  for (int idx = tid; idx < 80 * LDWWT; idx += 128) {
    int n = idx / LDWWT, k = idx % LDWWT;
    WwT_l[idx] = (n < 66 && k < NTM_C) ? Ww[k * 66 + n] : 0.f;
  }
  for (int idx = tid; idx < 16 * LDWFT; idx += 128) {
    int n = idx / LDWFT, k = idx % LDWFT;
    WfT_l[idx] = (n < NTM_O && k < NTM_C + NTM_V) ? Wf[k * NTM_O + n] : 0.f;
  }
  // ---- init recurrent state ----
  for (int idx = tid; idx < BB * NTM_N * NTM_V; idx += 128) M_l[idx] = Mb[idx % (NTM_N * NTM_V)];
  for (int idx = tid; idx < BB * NTM_N; idx += 128) wprev_l[idx] = 0.f;
  for (int idx = tid; idx < BB * NTM_V; idx += 128) r_l[idx] = rb[idx % NTM_V];
  for (int idx = tid; idx < BB * LDXR; idx += 128) xr_l[idx] = 0.f;
  for (int idx = tid; idx < BB * LDHR; idx += 128) hr_l[idx] = 0.f;
  if (tid < 112) bc_l[tid] = (tid < NTM_C) ? bc[tid] : 0.f;
  if (tid < 32) br_l[tid] = (tid < 26) ? br[tid] : 0.f;
  if (tid < 80) bw_l[tid] = (tid < 66) ? bw[tid] : 0.f;
  if (tid < 16) bf_l[tid] = (tid < NTM_O) ? bf[tid] : 0.f;
  __syncthreads();

  for (int t = 0; t < NTM_T; ++t) {
    // stage 1: xr = concat(x_t, r_prev)
    for (int idx = tid; idx < BB * NTM_I; idx += 128) {
      int j = idx / NTM_I, i = idx % NTM_I;
      xr_l[j * LDXR + i] = x[((size_t)(bb0 + j) * NTM_T + t) * NTM_I + i];
    }
    for (int idx = tid; idx < BB * NTM_V; idx += 128) {
      int j = idx / NTM_V, v = idx % NTM_V;
      xr_l[j * LDXR + NTM_I + v] = r_l[idx];
    }
    __syncthreads();

    // stage 2: controller h = tanh(xr@Wc + bc); 7 N-tiles over 4 waves (f32 WMMA)
    for (int tile = wave; tile < 7; tile += 4) {
      const int n0 = tile * 16;
      v8f acc = gemm_tile_lds(xr_l, LDXR, WcT_l, LDWCT, 32, n0, lane);
      const int half = lane >> 4, lr = lane & 15;
      const int col = n0 + lr;
#pragma unroll
      for (int v = 0; v < 8; ++v) {
        const int row = v + 8 * half;
        float val = (col < NTM_C) ? tanh_(acc[v] + bc_l[col]) : 0.f;
        h_l[row * LDH + col] = val;
        if (col < NTM_C) hr_l[row * LDHR + col] = val;
      }
    }
    __syncthreads();

    // stage 3: read-head (2 tiles) + write-head (5 tiles) parameter GEMMs
    for (int tile = wave; tile < 7; tile += 4) {
      const int half = lane >> 4, lr = lane & 15;
      if (tile < 2) {
        const int n0 = tile * 16;
        v8f acc = gemm_tile_lds(h_l, LDH, WrT_l, LDWRT, 112, n0, lane);
        const int col = n0 + lr;
#pragma unroll
        for (int v = 0; v < 8; ++v) pr_l[(v + 8 * half) * 32 + col] = acc[v] + br_l[col];
      } else {
        const int n0 = (tile - 2) * 16;
        v8f acc = gemm_tile_lds(h_l, LDH, WwT_l, LDWWT, 112, n0, lane);
        const int col = n0 + lr;
#pragma unroll
        for (int v = 0; v < 8; ++v) pw_l[(v + 8 * half) * 80 + col] = acc[v] + bw_l[col];
      }
    }
    __syncthreads();

    // stage 4: head activations
    for (int idx = tid; idx < BB * NTM_V; idx += 128) {
      int j = idx / NTM_V, v = idx % NTM_V;
      krh[idx] = tanh_(pr_l[j * 32 + v]);
      kwh[idx] = tanh_(pw_l[j * 80 + v]);
      eh[idx] = sigmoidf_(pw_l[j * 80 + 26 + v]);
      ah[idx] = tanh_(pw_l[j * 80 + 46 + v]);
    }
    __syncthreads();
    if (tid < BB) {
      const int j = tid;
      float nk = 0.f;
      for (int v = 0; v < NTM_V; ++v) { float kv = krh[j * NTM_V + v]; nk += kv * kv; }
      normkr[j] = sqrt_(nk);
      betar[j] = softplusf_(pr_l[j * 32 + 20]);
      gr[j] = sigmoidf_(pr_l[j * 32 + 21]);
      float e0 = pr_l[j * 32 + 22], e1 = pr_l[j * 32 + 23], e2 = pr_l[j * 32 + 24];
      float m = fmaxf(e0, fmaxf(e1, e2));
      float x0 = __expf(e0 - m), x1 = __expf(e1 - m), x2 = __expf(e2 - m);
      float si = rcp_(x0 + x1 + x2);
      srh[j * 3 + 0] = x0 * si; srh[j * 3 + 1] = x1 * si; srh[j * 3 + 2] = x2 * si;
      gammar[j] = 1.f + softplusf_(pr_l[j * 32 + 25]);
    } else if (tid < 2 * BB) {
      const int j = tid - BB;
      float nk = 0.f;
      for (int v = 0; v < NTM_V; ++v) { float kv = kwh[j * NTM_V + v]; nk += kv * kv; }
      normkw[j] = sqrt_(nk);
      betaw[j] = softplusf_(pw_l[j * 80 + 20]);
      gw[j] = sigmoidf_(pw_l[j * 80 + 21]);
      float e0 = pw_l[j * 80 + 22], e1 = pw_l[j * 80 + 23], e2 = pw_l[j * 80 + 24];
      float m = fmaxf(e0, fmaxf(e1, e2));
      float x0 = __expf(e0 - m), x1 = __expf(e1 - m), x2 = __expf(e2 - m);
      float si = rcp_(x0 + x1 + x2);
      srw[j * 3 + 0] = x0 * si; srw[j * 3 + 1] = x1 * si; srw[j * 3 + 2] = x2 * si;
      gammaw[j] = 1.f + softplusf_(pw_l[j * 80 + 25]);
    }
    __syncthreads();

    // stage 5: read addressing -> wr_l (uses previous w_w in wprev_l)
    ntm_address_(krh, betar, gr, srh, gammar, normkr, wprev_l, M_l,
                 sim_l, red_l, jt_l, wr_l, tid);

    // stage 6: r = w_r @ M ; splice into hr
    for (int idx = tid; idx < BB * NTM_V; idx += 128) {
      int j = idx / NTM_V, v = idx % NTM_V;
      const float* Mj = M_l + (size_t)j * NTM_N * NTM_V;
      const float* wj = wr_l + j * NTM_N;
      float acc = 0.f;
      for (int n = 0; n < NTM_N; ++n) acc += wj[n] * Mj[n * NTM_V + v];
      r_l[idx] = acc;
      hr_l[j * LDHR + NTM_C + v] = acc;
    }
    __syncthreads();

    // stage 7: output tile (wave 0 only): sigmoid(hr@Wf + bf) -> HBM
    if (wave == 0) {
      v8f acc = gemm_tile_lds(hr_l, LDHR, WfT_l, LDWFT, 128, 0, lane);
      const int half = lane >> 4, lr = lane & 15;
      if (lr < NTM_O) {
#pragma unroll
        for (int v = 0; v < 8; ++v) {
          const int row = v + 8 * half;
          out[((size_t)(bb0 + row) * NTM_T + t) * NTM_O + lr] = sigmoidf_(acc[v] + bf_l[lr]);
        }
      }
    }
    // prefetch next timestep's x while the write phase runs
    if (t + 1 < NTM_T && wave == 1 && lane < BB) {
      __builtin_prefetch(&x[((size_t)(bb0 + lane) * NTM_T + (t + 1)) * NTM_I], 0, 1);
    }

    // stage 8: write addressing -> wprev_l (this is w_w, carried to next step)
    ntm_address_(kwh, betaw, gw, srw, gammaw, normkw, wr_l, M_l,
                 sim_l, red_l, jt_l, wprev_l, tid);

    // stage 9: memory update M = M*(1 - w_w e) + w_w a
    for (int idx = tid; idx < BB * NTM_N * NTM_V; idx += 128) {
      int v = idx % NTM_V;
      int n = (idx / NTM_V) & (NTM_N - 1);
      int j = idx / (NTM_N * NTM_V);
      float wwv = wprev_l[j * NTM_N + n];
      float m = M_l[idx];
      M_l[idx] = m * (1.f - wwv * eh[j * NTM_V + v]) + wwv * ah[j * NTM_V + v];
    }
    __syncthreads();
  }
}

extern "C" void kernel_launch(void* const* d_in, const int* in_sizes, int n_in,
                              void* d_out, int out_size, void* d_ws, size_t ws_size,
                              hipStream_t stream) {
  (void)in_sizes; (void)n_in; (void)out_size; (void)d_ws; (void)ws_size;
  const float* x  = (const float*)d_in[0];
  const float* Wc = (const float*)d_in[1];
  const float* bc = (const float*)d_in[2];
  const float* Wr = (const float*)d_in[3];
  const float* br = (const float*)d_in[4];
  const float* Ww = (const float*)d_in[5];
  const float* bw = (const float*)d_in[6];
  const float* Wf = (const float*)d_in[7];
  const float* bf = (const float*)d_in[8];
  const float* rb = (const float*)d_in[9];
  const float* Mb = (const float*)d_in[10];
  float* out = (float*)d_out;

  dim3 grid(NTM_B / BB);   // 128 workgroups, one 16-batch tile each
  dim3 block(128);         // 4 wave32s
  ntm_fused_kernel<<<grid, block, 0, stream>>>(x, Wc, bc, Wr, br, Ww, bw, Wf, bf, rb, Mb, out);
}